// Sampler_1486058684695
// MI455X (gfx1250) — compile-verified
//
#include <hip/hip_runtime.h>
#include <stdint.h>

// MI455X (gfx1250) sampler, round 4:
//  - 3 streaming passes total: P1 (HBM) coarse hist+argmax, P2/P3 (L2-resident)
//    refine + fused survivor compaction; exact k-th-largest key resolved from
//    the compact LDS list (no 4th stream).
//  - all streams stage global data into LDS via double-buffered
//    global_load_async_to_lds_b128 + per-wave s_wait_asynccnt (lane-private slots)
//  - nucleus (top-p) select, Z, top-5 logprobs, Gumbel-max all run on the
//    compact list; q gathered only at survivor indices (~1K loads/row).
// HBM traffic ~65.6MB (logits once); 2 refinement passes hit the 192MB L2.

#define TPB   512
#define NW    (TPB / 32)
#define NB    4096
#define CAP   4096
#define NTOP  5
#define NEG_INF (-__builtin_inff())

__device__ __forceinline__ unsigned fkey(float x) {
    unsigned u = __float_as_uint(x);
    return u ^ ((u & 0x80000000u) ? 0xFFFFFFFFu : 0x80000000u);  // order-preserving
}

__device__ __forceinline__ void warpSumF(float& v) {
#pragma unroll
    for (int off = 16; off; off >>= 1) v += __shfl_xor(v, off, 32);
}
__device__ __forceinline__ void warpArgMax(float& v, int& i) {
#pragma unroll
    for (int off = 16; off; off >>= 1) {
        float ov = __shfl_xor(v, off, 32);
        int   oi = __shfl_xor(i, off, 32);
        if (ov > v || (ov == v && oi < i)) { v = ov; i = oi; }
    }
}
__device__ __forceinline__ void warpArgMax3(float& v, int& i, int& p) {
#pragma unroll
    for (int off = 16; off; off >>= 1) {
        float ov = __shfl_xor(v, off, 32);
        int   oi = __shfl_xor(i, off, 32);
        int   op = __shfl_xor(p, off, 32);
        if (ov > v || (ov == v && oi < i)) { v = ov; i = oi; p = op; }
    }
}

__global__ __launch_bounds__(TPB, 1) void sampler_kernel(
    const float* __restrict__ logits, const float* __restrict__ temperature,
    const int* __restrict__ top_k, const float* __restrict__ top_p,
    const float* __restrict__ q, float* __restrict__ out, int B, int V)
{
    __shared__ unsigned hist[NB];
    __shared__ float    whist[NB];
    __shared__ __align__(16) float ltile[2][TPB * 4];
    __shared__ float    sval[CAP];
    __shared__ int      sidx[CAP];
    __shared__ float    cval[TPB * NTOP];
    __shared__ int      cidx[TPB * NTOP];
    __shared__ float    rf[NW];
    __shared__ int      ri_[NW];
    __shared__ int      rp_[NW];
    __shared__ unsigned su_[NW];
    __shared__ float    sf_[NW];
    __shared__ int      s_selBin;
    __shared__ unsigned s_cumU;
    __shared__ float    s_cumF;
    __shared__ float    s_m, s_Z, s_Zf;
    __shared__ int      s_gi, s_ri, s_nsurv;
    __shared__ float    s_tv[NTOP];
    __shared__ int      s_ti[NTOP];

    const int b    = blockIdx.x;
    const int tid  = threadIdx.x;
    const int lane = tid & 31;
    const int wid  = tid >> 5;

    const float temp      = temperature[b];
    const bool  greedyRow = (temp < 1e-5f);
    const float invT      = greedyRow ? 1.0f : 1.0f / temp;
    int k = top_k[b]; if (k < 1) k = 1; if (k > V) k = V;
    const float tp = top_p[b];
    const float* lrow = logits + (size_t)b * V;
    const float* qrow = q      + (size_t)b * V;

    // ---- generic streaming pass: async global->LDS tiles, per-lane 16B slots.
    // Only per-wave s_wait_asynccnt needed (each lane consumes its own slot).
    auto stream_row = [&](const float* __restrict__ row, auto&& fn) {
        const int tileElems = TPB * 4;
        const int V4 = V & ~3;
        const int T  = (V4 + tileElems - 1) / tileElems;
        const int maxBase = V4 - 4;
        auto issue = [&](int t, int buf) {
            int base = t * tileElems + 4 * tid;
            if (base > maxBase) base = maxBase;  // clamp: duplicate load, ignored later
            unsigned ldsAddr = (unsigned)(size_t)&ltile[buf][tid * 4];
            unsigned long long gaddr = (unsigned long long)(size_t)(row + base);
            asm volatile("global_load_async_to_lds_b128 %0, %1, off"
                         :: "v"(ldsAddr), "v"(gaddr) : "memory");
        };
        if (T > 0 && maxBase >= 0) {
            issue(0, 0);
            for (int t = 0; t < T; ++t) {
                if (t + 1 < T) {
                    issue(t + 1, (t + 1) & 1);
                    asm volatile("s_wait_asynccnt 0x1" ::: "memory");
                } else {
                    asm volatile("s_wait_asynccnt 0x0" ::: "memory");
                }
                const int buf  = t & 1;
                const int base = t * tileElems + 4 * tid;
                if (base <= maxBase) {
#pragma unroll
                    for (int j = 0; j < 4; ++j) fn(base + j, ltile[buf][tid * 4 + j]);
                }
            }
        }
        for (int e = V4 + tid; e < V; e += TPB) fn(e, row[e]);  // tail
    };

    // ---- parallel select: find bin (descending) where cumulative-from-top crosses K ----
    auto selU = [&](const unsigned* bins, int nbins, unsigned K) {
        const int chunk = (nbins + TPB - 1) / TPB;
        int d0 = tid * chunk, d1 = d0 + chunk; if (d1 > nbins) d1 = nbins;
        unsigned csum = 0;
        for (int d = d0; d < d1; ++d) csum += bins[nbins - 1 - d];
        unsigned inc = csum;
#pragma unroll
        for (int off = 1; off < 32; off <<= 1) {
            unsigned o = __shfl_up(inc, off, 32);
            if (lane >= off) inc += o;
        }
        if (lane == 31) su_[wid] = inc;
        __syncthreads();
        if (tid == 0) {
            unsigned acc = 0;
            for (int w = 0; w < NW; ++w) { unsigned t0 = su_[w]; su_[w] = acc; acc += t0; }
            s_selBin = 0; s_cumU = acc;   // fallback: threshold never crossed
        }
        __syncthreads();
        const unsigned prefix = su_[wid] + inc - csum;
        if (prefix < K && prefix + csum >= K) {   // unique straddling thread
            unsigned cum = prefix;
            for (int d = d0; d < d1; ++d) {
                int bin = nbins - 1 - d;
                unsigned c = bins[bin];
                if (cum + c >= K) { s_selBin = bin; s_cumU = cum; break; }
                cum += c;
            }
        }
        __syncthreads();
    };
    auto selF = [&](const float* bins, int nbins, float K) {
        const int chunk = (nbins + TPB - 1) / TPB;
        int d0 = tid * chunk, d1 = d0 + chunk; if (d1 > nbins) d1 = nbins;
        float csum = 0.0f;
        for (int d = d0; d < d1; ++d) csum += bins[nbins - 1 - d];
        float inc = csum;
#pragma unroll
        for (int off = 1; off < 32; off <<= 1) {
            float o = __shfl_up(inc, off, 32);
            if (lane >= off) inc += o;
        }
        if (lane == 31) sf_[wid] = inc;
        __syncthreads();
        if (tid == 0) {
            float acc = 0.0f;
            for (int w = 0; w < NW; ++w) { float t0 = sf_[w]; sf_[w] = acc; acc += t0; }
            s_selBin = 0; s_cumF = acc;
        }
        __syncthreads();
        const float prefix = sf_[wid] + inc - csum;
        if (prefix < K && prefix + csum >= K) {
            float cum = prefix;
            for (int d = d0; d < d1; ++d) {
                int bin = nbins - 1 - d;
                float c = bins[bin];
                if (cum + c >= K) { s_selBin = bin; s_cumF = cum; break; }
                cum += c;
            }
        }
        __syncthreads();
    };

    // ---------- P1 (HBM pass): row argmax + coarse histogram (top 12 key bits) ----------
    for (int i = tid; i < NB; i += TPB) hist[i] = 0u;
    if (tid == 0) s_nsurv = 0;
    __syncthreads();
    float gv = NEG_INF; int gi = 0;
    stream_row(lrow, [&](int e, float v) {
        float x = v * invT;
        if (x > gv || (x == gv && e < gi)) { gv = x; gi = e; }
        atomicAdd(&hist[fkey(x) >> 20], 1u);
    });
    {   // block argmax -> s_m (row max) and s_gi (greedy sample)
        float av = gv; int ai = gi;
        warpArgMax(av, ai);
        if (lane == 0) { rf[wid] = av; ri_[wid] = ai; }
        __syncthreads();
        if (tid == 0) {
            float v = rf[0]; int id = ri_[0];
            for (int w = 1; w < NW; ++w)
                if (rf[w] > v || (rf[w] == v && ri_[w] < id)) { v = rf[w]; id = ri_[w]; }
            s_m = v; s_gi = id;
        }
        __syncthreads();
    }
    const float m = s_m;
    selU(hist, NB, (unsigned)k);
    const int      B1 = s_selBin;
    const unsigned r1 = (unsigned)k - s_cumU;   // rank within coarse bin B1

    // ---------- P2 (L2 pass): compact sure-survivors (bin > B1) + mid-bit hist of B1 ----------
    __syncthreads();
    for (int i = tid; i < NB; i += TPB) hist[i] = 0u;
    __syncthreads();
    stream_row(lrow, [&](int e, float v) {
        float x = v * invT;
        unsigned u = fkey(x);
        int c = (int)(u >> 20);
        if (c > B1) {                      // guaranteed survivor (< k of these)
            int slot = atomicAdd(&s_nsurv, 1);
            if (slot < CAP) { sval[slot] = x; sidx[slot] = e; }
        } else if (c == B1) {
            atomicAdd(&hist[(u >> 8) & 0xFFFu], 1u);
        }
    });
    __syncthreads();
    selU(hist, NB, r1);
    const int      B2 = s_selBin;
    const unsigned r2 = r1 - s_cumU;            // rank within (B1,B2) sub-bin

    // ---------- P3 (L2 pass): compact candidates (bin==B1, mid >= B2) ----------
    __syncthreads();
    for (int i = tid; i < 256; i += TPB) hist[i] = 0u;  // low-bit hist scratch (unused by P3 stream)
    __syncthreads();
    const unsigned pre24 = ((unsigned)B1 << 12) | (unsigned)B2;
    stream_row(lrow, [&](int e, float v) {
        float x = v * invT;
        unsigned u = fkey(x);
        if ((int)(u >> 20) == B1 && ((u >> 8) & 0xFFFu) >= (unsigned)B2) {
            int slot = atomicAdd(&s_nsurv, 1);
            if (slot < CAP) { sval[slot] = x; sidx[slot] = e; }
        }
    });
    __syncthreads();
    const int ns = (s_nsurv < CAP) ? s_nsurv : CAP;
    __syncthreads();

    // ---------- exact k-th-largest key from the list (LDS only, no 4th stream) ----------
    for (int i = tid; i < ns; i += TPB) {
        unsigned u = fkey(sval[i]);
        if ((u >> 8) == pre24) atomicAdd(&hist[u & 0xFFu], 1u);
    }
    __syncthreads();
    selU(hist, 256, r2);
    const unsigned Kk = (pre24 << 8) | (unsigned)s_selBin;

    // ---------- nucleus (top-p) threshold from the compact list (LDS only) ----------
    __syncthreads();
    for (int i = tid; i < NB; i += TPB) whist[i] = 0.0f;
    __syncthreads();
    float zloc = 0.0f;
    for (int i = tid; i < ns; i += TPB) {
        unsigned u = fkey(sval[i]);
        if (u >= Kk) {
            float w = __expf(sval[i] - m);
            zloc += w;
            atomicAdd(&whist[u >> 20], w);
        }
    }
    warpSumF(zloc);
    if (lane == 0) rf[wid] = zloc;
    __syncthreads();
    if (tid == 0) { float z = 0.0f; for (int w = 0; w < NW; ++w) z += rf[w]; s_Z = z; }
    __syncthreads();
    const float target = fmaxf(tp, 1e-12f) * s_Z;  // keep token iff mass strictly above < target
    selF(whist, NB, target);
    const int   Bp1      = s_selBin;
    const float cumAbove = s_cumF;

    __syncthreads();
    for (int i = tid; i < NB; i += TPB) whist[i] = 0.0f;
    __syncthreads();
    for (int i = tid; i < ns; i += TPB) {
        unsigned u = fkey(sval[i]);
        if (u >= Kk && (int)(u >> 20) == Bp1)
            atomicAdd(&whist[(u >> 8) & 0xFFFu], __expf(sval[i] - m));
    }
    __syncthreads();
    selF(whist, NB, target - cumAbove);
    const unsigned Kp = ((unsigned)Bp1 << 20) | ((unsigned)s_selBin << 8);
    const unsigned Kf = (Kp > Kk) ? Kp : Kk;
    __syncthreads();

    // ---------- final: Z, Gumbel-max argmax, top-5 -- all from the list ----------
    // q touched only at survivor indices (tiny gather instead of 65MB stream).
    float Zf = 0.0f;
    float rv = NEG_INF; int rix = 0;
    float tv[NTOP]; int ti[NTOP];
#pragma unroll
    for (int s = 0; s < NTOP; ++s) { tv[s] = NEG_INF; ti[s] = 0; }

    for (int i = tid; i < ns; i += TPB) {
        float x = sval[i];
        if (fkey(x) >= Kf) {
            int   e = sidx[i];
            float w = __expf(x - m);
            Zf += w;
            float ee = -__logf(qrow[e]);
            float r  = w / ee;
            if (r > rv || (r == rv && e < rix)) { rv = r; rix = e; }
            if (x > tv[NTOP - 1]) {
                float vv = x; int id = e;
#pragma unroll
                for (int s = 0; s < NTOP; ++s) {
                    if (vv > tv[s]) { float t0 = tv[s]; int i0 = ti[s]; tv[s] = vv; ti[s] = id; vv = t0; id = i0; }
                }
            }
        }
    }

    // ---- block reductions ----
    warpSumF(Zf);
    if (lane == 0) rf[wid] = Zf;
    __syncthreads();
    if (tid == 0) { float z = 0.0f; for (int w = 0; w < NW; ++w) z += rf[w]; s_Zf = z; }
    __syncthreads();

    warpArgMax(rv, rix);
    if (lane == 0) { rf[wid] = rv; ri_[wid] = rix; }
    __syncthreads();
    if (tid == 0) {
        float v = rf[0]; int id = ri_[0];
        for (int w = 1; w < NW; ++w) if (rf[w] > v || (rf[w] == v && ri_[w] < id)) { v = rf[w]; id = ri_[w]; }
        s_ri = id;
    }
    __syncthreads();

    // ---- global top-5 merge from per-thread candidates ----
#pragma unroll
    for (int s = 0; s < NTOP; ++s) { cval[tid * NTOP + s] = tv[s]; cidx[tid * NTOP + s] = ti[s]; }
    __syncthreads();
    for (int s = 0; s < NTOP; ++s) {
        float bv = NEG_INF; int bi = 0x7FFFFFFF; int bp = -1;
        for (int c = tid; c < TPB * NTOP; c += TPB) {
            float v = cval[c]; int id = cidx[c];
            if (v > bv || (v == bv && id < bi)) { bv = v; bi = id; bp = c; }
        }
        warpArgMax3(bv, bi, bp);
        if (lane == 0) { rf[wid] = bv; ri_[wid] = bi; rp_[wid] = bp; }
        __syncthreads();
        if (tid == 0) {
            float v = rf[0]; int id = ri_[0]; int p = rp_[0];
            for (int w = 1; w < NW; ++w)
                if (rf[w] > v || (rf[w] == v && ri_[w] < id)) { v = rf[w]; id = ri_[w]; p = rp_[w]; }
            s_tv[s] = v; s_ti[s] = id;
            if (p >= 0) cval[p] = NEG_INF;
        }
        __syncthreads();
    }

    // ---- outputs: [sampled i32 xB | logprobs f32 x5B | indices i32 x5B] ----
    if (tid == 0) {
        int* iout = (int*)out;
        iout[b] = greedyRow ? s_gi : s_ri;
        float logZ = __logf(s_Zf);
#pragma unroll
        for (int s = 0; s < NTOP; ++s) {
            float v  = s_tv[s];
            float lp = (v == NEG_INF) ? NEG_INF : (v - m - logZ);
            out[B + b * NTOP + s] = lp;
            iout[B + B * NTOP + b * NTOP + s] = s_ti[s];
        }
    }
}

extern "C" void kernel_launch(void* const* d_in, const int* in_sizes, int n_in,
                              void* d_out, int out_size, void* d_ws, size_t ws_size,
                              hipStream_t stream) {
    const float* logits      = (const float*)d_in[0];
    const float* temperature = (const float*)d_in[1];
    const int*   top_k       = (const int*)d_in[2];
    const float* top_p       = (const float*)d_in[3];
    const float* q           = (const float*)d_in[4];
    (void)d_ws; (void)ws_size; (void)out_size; (void)n_in;

    const int B = in_sizes[1];
    const int V = in_sizes[0] / B;

    sampler_kernel<<<dim3(B), dim3(TPB), 0, stream>>>(
        logits, temperature, top_k, top_p, q, (float*)d_out, B, V);
}